// TFF_Angle_76020921139246
// MI455X (gfx1250) — compile-verified
//
#include <hip/hip_runtime.h>

// CDNA5 wave32 vector types for the fp32 WMMA reduction.
typedef __attribute__((ext_vector_type(2))) float v2f;  // A/B of 16x16x4 f32: 64 vals / 32 lanes
typedef __attribute__((ext_vector_type(8))) float v8f;  // C/D 16x16 f32: 256 vals / 32 lanes

__device__ __forceinline__ void gAdd(float* p, float v) {
    // Lowers to GLOBAL_ATOMIC_ADD_F32 (no CAS loop).
    unsafeAtomicAdd(p, v);
}

// d_out layout: [0] = energy, [1 .. 1+12287] = forces (4096 x 3).
__global__ void __launch_bounds__(256) tff_angle_init(float* __restrict__ out,
                                                      const float* __restrict__ forces_in,
                                                      int nF) {
    int t = blockIdx.x * blockDim.x + threadIdx.x;
    if (t == 0) out[0] = 0.0f;
    if (t < nF) out[1 + t] = forces_in[t];
}

__global__ void __launch_bounds__(256) tff_angle_kernel(
        const float* __restrict__ dist,     // 4096 x 4096
        const float* __restrict__ vec,      // 4096 x 4096 x 3
        const float* __restrict__ params,   // N x 2 (k0, theta0)
        const int*   __restrict__ cidx,     // N x 3
        const int*   __restrict__ pCalcE,
        const int*   __restrict__ pCalcF,
        float* __restrict__ out,
        int nAngles, int totalThreads) {
    const int tid   = blockIdx.x * blockDim.x + threadIdx.x;
    const int calcE = *pCalcE;   // uniform scalar loads
    const int calcF = *pCalcF;

    float e0 = 0.0f, e1 = 0.0f;

    // Two angles per thread, fully unrolled: all gathers issued up front for MLP.
    // No early returns anywhere: EXEC must be all-1s at the WMMA below.
    #pragma unroll
    for (int it = 0; it < 2; ++it) {
        const int  i     = tid + it * totalThreads;
        const bool valid = (i < nAngles);
        const int  ii    = valid ? i : 0;   // clamp to a safe address; contribution masked below

        const int a1 = cidx[3 * ii + 0];
        const int a2 = cidx[3 * ii + 1];
        const int a3 = cidx[3 * ii + 2];
        const float2 kp = ((const float2*)params)[ii];   // {k0, theta0}, one b64 load

        const size_t r21 = ((size_t)a2 << 12) + (size_t)a1;   // a2*4096 + a1
        const size_t r23 = ((size_t)a2 << 12) + (size_t)a3;
        const float* v21p = vec + r21 * 3u;
        const float* v23p = vec + r23 * 3u;

        const float v21x = v21p[0], v21y = v21p[1], v21z = v21p[2];
        const float v23x = v23p[0], v23y = v23p[1], v23z = v23p[2];
        const float d21  = dist[r21];
        const float d23  = dist[r23];

        float cosT  = fminf(1.0f, fmaxf(-1.0f, v21x * v23x + v21y * v23y + v21z * v23z));
        float theta = acosf(cosT);
        float dt    = theta - kp.y;
        float ei    = valid ? (kp.x * dt * dt) : 0.0f;
        if (it == 0) e0 = ei; else e1 = ei;

        if (calcF && valid) {   // calcF uniform; valid diverges only on the padded tail
            float sinT = sqrtf(1.0f - cosT * cosT);
            float coef = (sinT > 0.0f) ? (-2.0f * kp.x * dt / fmaxf(sinT, 1e-12f)) : 0.0f;
            float c21 = coef / d21;
            float c23 = coef / d23;

            float f0x = (cosT * v21x - v23x) * c21;
            float f0y = (cosT * v21y - v23y) * c21;
            float f0z = (cosT * v21z - v23z) * c21;
            float f2x = (cosT * v23x - v21x) * c23;
            float f2y = (cosT * v23y - v21y) * c23;
            float f2z = (cosT * v23z - v21z) * c23;

            float* f = out + 1;
            gAdd(f + 3 * a1 + 0, f0x);
            gAdd(f + 3 * a1 + 1, f0y);
            gAdd(f + 3 * a1 + 2, f0z);
            gAdd(f + 3 * a2 + 0, -(f0x + f2x));
            gAdd(f + 3 * a2 + 1, -(f0y + f2y));
            gAdd(f + 3 * a2 + 2, -(f0z + f2z));
            gAdd(f + 3 * a3 + 0, f2x);
            gAdd(f + 3 * a3 + 1, f2y);
            gAdd(f + 3 * a3 + 2, f2z);
        }
    }

    if (calcE) {   // uniform branch; EXEC all-1s here (required for WMMA)
        // Wave-level reduction on the matrix pipe:
        // A(16x4 f32) = this wave's 64 energy terms (2 VGPRs/lane), B = ones.
        // D[m][n] = sum_k A[m][k]  -> every N-column replicates the row sums,
        // so sum(all D elements) == 16 * sum(e).
        v2f A; A[0] = e0;   A[1] = e1;
        v2f B; B[0] = 1.0f; B[1] = 1.0f;
        v8f C = {};
        C = __builtin_amdgcn_wmma_f32_16x16x4_f32(
                /*neg_a=*/false, A, /*neg_b=*/false, B,
                /*c_mod=*/(short)0, C, /*reuse_a=*/false, /*reuse_b=*/false);

        float s = 0.0f;
        #pragma unroll
        for (int j = 0; j < 8; ++j) s += C[j];

        __shared__ float sE;
        if (threadIdx.x == 0) sE = 0.0f;
        __syncthreads();
        atomicAdd(&sE, s);              // ds_add_f32
        __syncthreads();
        if (threadIdx.x == 0) gAdd(out, sE * 0.0625f);   // /16 for column replication
    }
}

extern "C" void kernel_launch(void* const* d_in, const int* in_sizes, int n_in,
                              void* d_out, int out_size, void* d_ws, size_t ws_size,
                              hipStream_t stream) {
    const float* dist   = (const float*)d_in[0];   // dist_mat
    const float* vec    = (const float*)d_in[1];   // vector_mat
    const float* fin    = (const float*)d_in[2];   // forces_out (zeros)
    const float* params = (const float*)d_in[3];   // (N,2)
    const int*   cidx   = (const int*)  d_in[4];   // (N,3)
    const int*   pCE    = (const int*)  d_in[5];
    const int*   pCF    = (const int*)  d_in[6];
    float*       out    = (float*)d_out;

    const int nAngles = in_sizes[4] / 3;
    const int nF      = in_sizes[2];               // 4096*3

    // Re-init output every call (harness poisons d_out; forces start from forces_out).
    tff_angle_init<<<(nF + 255) / 256, 256, 0, stream>>>(out, fin, nF);

    if (nAngles <= 0) return;
    const int threadsNeeded = (nAngles + 1) / 2;   // 2 angles per thread
    const int blocks        = (threadsNeeded + 255) / 256;
    const int totalThreads  = blocks * 256;
    tff_angle_kernel<<<blocks, 256, 0, stream>>>(dist, vec, params, cidx,
                                                 pCE, pCF, out, nAngles, totalThreads);
}